// ComplexRnn_6287832121961
// MI455X (gfx1250) — compile-verified
//
#include <hip/hip_runtime.h>
#include <math.h>

typedef __attribute__((ext_vector_type(2))) float v2f;
typedef __attribute__((ext_vector_type(8))) float v8f;

#define NC   5
#define JDIM 4096
#define KDIM 4096
#define JK   ((size_t)JDIM * (size_t)KDIM)   // 16,777,216 elements per slice
#define EPSN 1e-12f

__device__ __forceinline__ float hw_tanh(float x) {
  float r;
  asm("v_tanh_f32 %0, %1" : "=v"(r) : "v"(x));
  return r;
}

// Combined coefficient matrix M[c, kk]:
//   kk in [0,5):  K_slow[kk, c]   (einsum 'ic,ijk->cjk' => transpose access)
//   kk in [5,10): Q[kk-5, c]
//   else 0; rows c >= 5 are zero.
__device__ __forceinline__ float mcoef(const float* Ks, const float* Q, int row, int kk) {
  if (row >= NC) return 0.0f;
  if (kk < NC)      return Ks[kk * NC + row];
  if (kk < 2 * NC)  return Q[(kk - NC) * NC + row];
  return 0.0f;
}

// ---------------------------------------------------------------------------
// Pass 1: pre = M @ [chem; chem_fast] via 3x V_WMMA_F32_16X16X4_F32,
//         t = y*chem + z*tanh(pre), write unnormalized new_chem,
//         accumulate sum_j t^2 into sumsq[c, k].
// Wave layout: lane n = lane&15 -> k column; hf = lane>>4 selects which
// B-matrix rows this lane's registers carry (B f32 4x16: row kk -> vgpr kk%2,
// lane-half kk/2; A 16x4: lane m holds A[m, 2*hf + vgpr]).
// All addressing: uniform SGPR slice base + 32-bit element offset (GVS form).
// ---------------------------------------------------------------------------
__global__ void __launch_bounds__(256) pass1_wmma(
    const float* __restrict__ chem, const float* __restrict__ chemf,
    const float* __restrict__ Ks,   const float* __restrict__ Q,
    const float* __restrict__ yv,   const float* __restrict__ zv,
    float* __restrict__ out_chem,   float* __restrict__ sumsq, int j_chunk) {
  const int tid  = threadIdx.x;
  const int lane = tid & 31;
  const int wave = tid >> 5;
  const int n    = lane & 15;   // B column / output position within wave
  const int hf   = lane >> 4;   // lane half
  const int k    = blockIdx.x * 128 + wave * 16 + n;
  const int j0   = blockIdx.y * j_chunk;
  const int m    = n;           // A row carried by this lane

  v2f a0, a1, a2;
  a0.x = mcoef(Ks, Q, m, 0 + 2 * hf);  a0.y = mcoef(Ks, Q, m, 1 + 2 * hf);
  a1.x = mcoef(Ks, Q, m, 4 + 2 * hf);  a1.y = mcoef(Ks, Q, m, 5 + 2 * hf);
  a2.x = mcoef(Ks, Q, m, 8 + 2 * hf);  a2.y = mcoef(Ks, Q, m, 9 + 2 * hf);

  float yc[NC], zc[NC], acc[NC];
#pragma unroll
  for (int c = 0; c < NC; ++c) { yc[c] = yv[c]; zc[c] = zv[c]; acc[c] = 0.0f; }

  for (int jj = 0; jj < j_chunk; ++jj) {
    const unsigned p = (unsigned)((j0 + jj) * KDIM + k);  // < 2^24, GVS offset
    float cc[NC], ff[NC];
#pragma unroll
    for (int i = 0; i < NC; ++i) {
      cc[i] = __builtin_nontemporal_load((chem  + (size_t)i * JK) + p);
      ff[i] = __builtin_nontemporal_load((chemf + (size_t)i * JK) + p);
    }
    // B rows (global K order): 0..4 = chem0..4, 5..9 = chemf0..4, 10..11 pad.
    v2f b0, b1, b2;
    b0.x = hf ? cc[2] : cc[0];  b0.y = hf ? cc[3] : cc[1];
    b1.x = hf ? ff[1] : cc[4];  b1.y = hf ? ff[2] : ff[0];
    b2.x = ff[3];               b2.y = ff[4];   // hi-half rows hit zero A cols

    v8f d = {};
    d = __builtin_amdgcn_wmma_f32_16x16x4_f32(false, a0, false, b0, (short)0, d, false, false);
    d = __builtin_amdgcn_wmma_f32_16x16x4_f32(false, a1, false, b1, (short)0, d, false, false);
    d = __builtin_amdgcn_wmma_f32_16x16x4_f32(false, a2, false, b2, (short)0, d, false, false);

    float t[NC];
#pragma unroll
    for (int c = 0; c < NC; ++c) {
      t[c] = yc[c] * cc[c] + zc[c] * hw_tanh(d[c]);
      acc[c] += t[c] * t[c];
    }
    if (hf == 0) {   // single exec toggle; 5 stores can clause together
#pragma unroll
      for (int c = 0; c < NC; ++c)
        __builtin_nontemporal_store(t[c], (out_chem + (size_t)c * JK) + p);
    }
  }
  if (hf == 0) {
#pragma unroll
    for (int c = 0; c < NC; ++c)
      atomicAdd(&sumsq[c * KDIM + k], acc[c]);
  }
}

// ---------------------------------------------------------------------------
// Pass 2: normalize new_chem over j, mix with softmax(v) -> new_value
//         (unnormalized), accumulate sum_j new_value^2 into colsq[k].
// ---------------------------------------------------------------------------
__global__ void __launch_bounds__(256) pass2_norm_mix(
    const float* __restrict__ vvec, float* __restrict__ out_chem,
    float* __restrict__ out_val, const float* __restrict__ sumsq,
    float* __restrict__ colsq, int j_chunk) {
  const int k  = blockIdx.x * blockDim.x + threadIdx.x;
  const int j0 = blockIdx.y * j_chunk;

  float vs[NC], vmax = -3.0e38f;
#pragma unroll
  for (int c = 0; c < NC; ++c) { vs[c] = vvec[c]; vmax = fmaxf(vmax, vs[c]); }
  float se = 0.0f;
#pragma unroll
  for (int c = 0; c < NC; ++c) { vs[c] = __expf(vs[c] - vmax); se += vs[c]; }
  const float sinv = 1.0f / se;
#pragma unroll
  for (int c = 0; c < NC; ++c) vs[c] *= sinv;

  float ninv[NC];
#pragma unroll
  for (int c = 0; c < NC; ++c)
    ninv[c] = 1.0f / fmaxf(sqrtf(sumsq[c * KDIM + k]), EPSN);

  float accv = 0.0f;
  for (int jj = 0; jj < j_chunk; ++jj) {
    const unsigned p = (unsigned)((j0 + jj) * KDIM + k);
    float val = 0.0f;
#pragma unroll
    for (int c = 0; c < NC; ++c) {
      float* oc = out_chem + (size_t)c * JK;
      float t = oc[p] * ninv[c];
      oc[p] = t;
      val += vs[c] * t;
    }
    out_val[p] = val;
    accv += val * val;
  }
  atomicAdd(&colsq[k], accv);
}

// ---------------------------------------------------------------------------
// Pass 3: normalize new_value columns over j.
// ---------------------------------------------------------------------------
__global__ void __launch_bounds__(256) pass3_colnorm(
    float* __restrict__ out_val, const float* __restrict__ colsq, int j_chunk) {
  const int k  = blockIdx.x * blockDim.x + threadIdx.x;
  const int j0 = blockIdx.y * j_chunk;
  const float ninv = 1.0f / fmaxf(sqrtf(colsq[k]), EPSN);
  for (int jj = 0; jj < j_chunk; ++jj) {
    const unsigned p = (unsigned)((j0 + jj) * KDIM + k);
    out_val[p] *= ninv;
  }
}

__global__ void zero_ws(float* w, int nTot) {
  int i = blockIdx.x * blockDim.x + threadIdx.x;
  if (i < nTot) w[i] = 0.0f;
}

extern "C" void kernel_launch(void* const* d_in, const int* in_sizes, int n_in,
                              void* d_out, int out_size, void* d_ws, size_t ws_size,
                              hipStream_t stream) {
  const float* chem  = (const float*)d_in[0];
  const float* chemf = (const float*)d_in[1];
  const float* Ks    = (const float*)d_in[2];
  const float* Q     = (const float*)d_in[3];
  const float* yv    = (const float*)d_in[4];
  const float* zv    = (const float*)d_in[5];
  const float* vvec  = (const float*)d_in[6];

  float* out_chem = (float*)d_out;                 // [5, 4096, 4096]
  float* out_val  = out_chem + (size_t)NC * JK;    // [4096, 4096]

  float* sumsq = (float*)d_ws;                     // [5, 4096]
  float* colsq = sumsq + NC * KDIM;                // [4096]

  const int nz = NC * KDIM + KDIM;
  zero_ws<<<(nz + 255) / 256, 256, 0, stream>>>(sumsq, nz);

  const int JCH = 128;                             // j per block => 32 j-blocks
  dim3 g1(KDIM / 128, JDIM / JCH);                 // (32, 32), 8 waves x 16 k
  pass1_wmma<<<g1, 256, 0, stream>>>(chem, chemf, Ks, Q, yv, zv,
                                     out_chem, sumsq, JCH);

  dim3 g2(KDIM / 256, JDIM / JCH);                 // (16, 32)
  pass2_norm_mix<<<g2, 256, 0, stream>>>(vvec, out_chem, out_val,
                                         sumsq, colsq, JCH);

  dim3 g3(KDIM / 256, JDIM / JCH);                 // (16, 32)
  pass3_colnorm<<<g3, 256, 0, stream>>>(out_val, colsq, JCH);
}